// PerfectTreeTraversalDecisionTreeImplKeras_37744172597272
// MI455X (gfx1250) — compile-verified
//
#include <hip/hip_runtime.h>

// Problem constants (from reference): B=16384, F=256, T=512, D=7, C=4
#define NF 256
#define NT 512
#define ND 7
#define NODES_TOTAL (NT * ((1 << (ND + 1)) - 2)) // 130048

typedef __attribute__((ext_vector_type(2))) float v2f;
typedef __attribute__((ext_vector_type(8))) float v8f;

// --- Pre-pass: fuse (node, bias) SoA -> int2 AoS, and (root_node, root_bias,
// tree_index) -> int4, so the traversal needs ONE gather per level, not two.
__global__ __launch_bounds__(256) void pack_kernel(
    const int* __restrict__ nodes, const float* __restrict__ biases,
    const int* __restrict__ root_nodes, const float* __restrict__ root_biases,
    const int* __restrict__ tree_indices,
    int2* __restrict__ pn, int4* __restrict__ pr) {
  int i = blockIdx.x * 256 + threadIdx.x;
  if (i < NODES_TOTAL) pn[i] = make_int2(nodes[i], __float_as_int(biases[i]));
  if (i < NT)
    pr[i] = make_int4(root_nodes[i], __float_as_int(root_biases[i]),
                      tree_indices[i], 0);
}

// --- Main traversal: one block per sample, 256 threads, 2 trees per thread
// (two independent software-pipelined chains). x row staged in LDS (async on
// gfx1250). PACKED path speculatively loads the child-pair for level l+1 as a
// single aligned b128 one level ahead, hiding L2 gather latency.
template <bool PACKED>
__global__ __launch_bounds__(256) void traverse_kernel(
    const float* __restrict__ x,
    const int2* __restrict__ pn, const int4* __restrict__ pr,
    const int* __restrict__ nodes, const float* __restrict__ biases,
    const int* __restrict__ root_nodes, const float* __restrict__ root_biases,
    const int* __restrict__ tree_indices,
    const float4* __restrict__ leaves, float4* __restrict__ out) {
  __shared__ float xs[NF];
  __shared__ float4 buf1[256];
  __shared__ float4 buf2[32];

  const int tid = threadIdx.x;
  const int b = blockIdx.x;

  // Stage this sample's feature row into LDS (256 floats, 1 per thread).
#if __has_builtin(__builtin_amdgcn_global_load_async_to_lds_b32)
  __builtin_amdgcn_global_load_async_to_lds_b32(
      (__attribute__((address_space(1))) int*)(x + (size_t)b * NF + tid),
      (__attribute__((address_space(3))) int*)&xs[tid], 0, 0);
  asm volatile("s_wait_asynccnt 0x0" ::: "memory");
#else
  xs[tid] = x[(size_t)b * NF + tid];
#endif
  __syncthreads();

  const int t0 = tid;
  const int t1 = tid + 256;

  float4 lf0, lf1;

  if constexpr (PACKED) {
    // Pair view: pn4[k] = {node[2k], node[2k+1]} of the packed level arrays.
    const int4* __restrict__ pn4 = (const int4*)pn;

    // Level-1 pairs cover indices {2t, 2t+1}; issue before the root compare.
    int4 p0 = pn4[t0];
    int4 p1 = pn4[t1];
    int4 r0 = pr[t0];
    int4 r1 = pr[t1];
    int i0 = r0.z + ((xs[r0.x] < __int_as_float(r0.y)) ? 1 : 0);
    int i1 = r1.z + ((xs[r1.x] < __int_as_float(r1.y)) ? 1 : 0);

    float4 la0, lb0, la1, lb1;
#pragma unroll
    for (int l = 1; l <= ND; ++l) {
      int4 np0, np1;
      if (l < ND) {
        // pair for level l+1: covers {2*i_{l-1}, 2*i_{l-1}+1}; needs only the
        // CURRENT index, so it overlaps the LDS gather + compare below.
        const int offq = NT * ((1 << l) - 1);
        np0 = pn4[offq + i0];
        np1 = pn4[offq + i1];
      } else {
        // speculatively fetch both candidate leaves (2 x b128, contiguous)
        la0 = leaves[2 * i0];
        lb0 = leaves[2 * i0 + 1];
        la1 = leaves[2 * i1];
        lb1 = leaves[2 * i1 + 1];
      }
      int2 n0 = (i0 & 1) ? make_int2(p0.z, p0.w) : make_int2(p0.x, p0.y);
      int2 n1 = (i1 & 1) ? make_int2(p1.z, p1.w) : make_int2(p1.x, p1.y);
      i0 = 2 * i0 + ((xs[n0.x] < __int_as_float(n0.y)) ? 1 : 0);
      i1 = 2 * i1 + ((xs[n1.x] < __int_as_float(n1.y)) ? 1 : 0);
      p0 = np0;
      p1 = np1;
    }
    lf0 = (i0 & 1) ? lb0 : la0;
    lf1 = (i1 & 1) ? lb1 : la1;
  } else {
    int i0 = tree_indices[t0] + ((xs[root_nodes[t0]] < root_biases[t0]) ? 1 : 0);
    int i1 = tree_indices[t1] + ((xs[root_nodes[t1]] < root_biases[t1]) ? 1 : 0);
    int off = 0;
#pragma unroll
    for (int l = 1; l <= ND; ++l) {
      int f0 = nodes[off + i0];
      int f1 = nodes[off + i1];
      float b0 = biases[off + i0];
      float b1 = biases[off + i1];
      i0 = 2 * i0 + ((xs[f0] < b0) ? 1 : 0);
      i1 = 2 * i1 + ((xs[f1] < b1) ? 1 : 0);
      off += (NT << l);
    }
    lf0 = leaves[i0];
    lf1 = leaves[i1];
  }

  float4 acc;
  acc.x = lf0.x + lf1.x;
  acc.y = lf0.y + lf1.y;
  acc.z = lf0.z + lf1.z;
  acc.w = lf0.w + lf1.w;

  // ---- Block reduction (low DS-op count; wave32-aware; exact f32 WMMA) ----
  buf1[tid] = acc; // one b128 store per thread
  __syncthreads();

  if (tid < 32) {
    // Each lane sums 8 threads' float4 (8 x ds_load_b128, independent).
    float4 s = buf1[tid * 8];
#pragma unroll
    for (int k = 1; k < 8; ++k) {
      float4 t = buf1[tid * 8 + k];
      s.x += t.x;
      s.y += t.y;
      s.z += t.z;
      s.w += t.w;
    }
    // Transpose through LDS: flat[L*4+c] = s[L][c]  (per-wave in-order DS).
    buf2[tid] = s;
    const float* bf = (const float*)buf2;
    // v[L] = sum over q in {m, m+8, m+16, m+24} of s[q][c], m=L>>2, c=L&3.
    float v = bf[tid] + bf[tid + 32] + bf[tid + 64] + bf[tid + 96];

    float s0, s1, s2, s3;
#if __has_builtin(__builtin_amdgcn_wmma_f32_16x16x4_f32)
    // Exact f32 WMMA row-sum: B = ones (layout-invariant), C = 0 =>
    // D[m][n] = sum_k A[m][k].  A layout (documented): V0 = {m:K0 | m:K2},
    // V1 = {K1 | K3}; with A1 = 0, rowsum(m) = v[m] + v[m+16].
    v2f A;
    A[0] = v;
    A[1] = 0.0f;
    v2f Bones;
    Bones[0] = 1.0f;
    Bones[1] = 1.0f;
    v8f Cm = {};
    v8f Dv = __builtin_amdgcn_wmma_f32_16x16x4_f32(
        false, A, false, Bones, (short)0, Cm, false, false);
    // lane 0..15: Dv[m] = rows 0..7; lane16..31: rows 8..15.
    // out[c] = sum_{L = 4m+c} v[L] = (lane0: d[c]+d[c+4]) + (lane16: same)
    s0 = Dv[0] + Dv[4];
    s1 = Dv[1] + Dv[5];
    s2 = Dv[2] + Dv[6];
    s3 = Dv[3] + Dv[7];
    s0 += __shfl_xor(s0, 16, 32);
    s1 += __shfl_xor(s1, 16, 32);
    s2 += __shfl_xor(s2, 16, 32);
    s3 += __shfl_xor(s3, 16, 32);
#else
    s0 = s1 = s2 = s3 = 0.0f;
#pragma unroll
    for (int q = 0; q < 32; ++q) {
      s0 += bf[q * 4 + 0];
      s1 += bf[q * 4 + 1];
      s2 += bf[q * 4 + 2];
      s3 += bf[q * 4 + 3];
    }
    (void)v;
#endif
    if (tid == 0) out[b] = make_float4(s0, s1, s2, s3);
  }
}

extern "C" void kernel_launch(void* const* d_in, const int* in_sizes, int n_in,
                              void* d_out, int out_size, void* d_ws,
                              size_t ws_size, hipStream_t stream) {
  const float* x = (const float*)d_in[0];
  const int* root_nodes = (const int*)d_in[1];
  const float* root_biases = (const float*)d_in[2];
  const int* tree_indices = (const int*)d_in[3];
  const int* nodes_flat = (const int*)d_in[4];
  const float* biases_flat = (const float*)d_in[5];
  const float4* leaves = (const float4*)d_in[6];
  float4* out = (float4*)d_out;

  const int nb = in_sizes[0] / NF; // number of samples (16384)

  const size_t pn_bytes = (size_t)NODES_TOTAL * sizeof(int2); // 16B-multiple
  const size_t need = pn_bytes + (size_t)NT * sizeof(int4);

  if (ws_size >= need) {
    int2* pn = (int2*)d_ws;
    int4* pr = (int4*)((char*)d_ws + pn_bytes);
    pack_kernel<<<(NODES_TOTAL + 255) / 256, 256, 0, stream>>>(
        nodes_flat, biases_flat, root_nodes, root_biases, tree_indices, pn, pr);
    traverse_kernel<true><<<nb, 256, 0, stream>>>(
        x, pn, pr, nullptr, nullptr, nullptr, nullptr, nullptr, leaves, out);
  } else {
    traverse_kernel<false><<<nb, 256, 0, stream>>>(
        x, nullptr, nullptr, nodes_flat, biases_flat, root_nodes, root_biases,
        tree_indices, leaves, out);
  }
}